// Model_46815143526465
// MI455X (gfx1250) — compile-verified
//
#include <hip/hip_runtime.h>
#include <math.h>

#define N_NODES 100000
#define N_EDGES 1600000
#define N_GRAPHS 200
#define NPG 500
#define K_SORT 50
#define NFEAT 64
#define HID 32
#define CAT 96
#define NCLASS 10

typedef __attribute__((ext_vector_type(2))) float v2f;
typedef __attribute__((ext_vector_type(8))) float v8f;

// ---------------- degree / gcn-norm ----------------
__global__ void k_init_deg(float* deg) {
  int i = blockIdx.x * blockDim.x + threadIdx.x;
  if (i < N_NODES) deg[i] = 1.0f;   // self-loop
}

__global__ void k_deg_edge(float* deg, const int* __restrict__ dst) {
  int e = blockIdx.x * blockDim.x + threadIdx.x;
  if (e < N_EDGES) atomicAdd(&deg[dst[e]], 1.0f);
}

__global__ void k_dinv(float* dinv) {
  int i = blockIdx.x * blockDim.x + threadIdx.x;
  if (i < N_NODES) dinv[i] = rsqrtf(dinv[i]);
}

// ---------------- SGConv propagation ----------------
// xout[n,f] = dinv[n]^2 * xin[n,f]   (self contribution)
__global__ void k_prop_init(const float* __restrict__ xin, float* __restrict__ xout,
                            const float* __restrict__ dinv, int d) {
  int i = blockIdx.x * blockDim.x + threadIdx.x;
  int nd = N_NODES * d;
  if (i < nd) {
    int n = i / d;
    float w = dinv[n];
    xout[i] = w * w * xin[i];
  }
}

// xout[dst] += dinv[src]*dinv[dst] * xin[src]   (4 features per thread, coalesced)
__global__ void k_prop_edge(const float* __restrict__ xin, float* __restrict__ xout,
                            const int* __restrict__ src, const int* __restrict__ dst,
                            const float* __restrict__ dinv, int d4) {
  int i = blockIdx.x * blockDim.x + threadIdx.x;
  int tot = N_EDGES * d4;
  if (i >= tot) return;
  int e = i / d4, f4 = i - e * d4;
  int s = src[e], t = dst[e];
  float w = dinv[s] * dinv[t];
  const float4* xi = (const float4*)xin;
  float4 v = xi[(long)s * d4 + f4];
  float* o = xout + ((long)t * d4 + f4) * 4;
  atomicAdd(o + 0, w * v.x);
  atomicAdd(o + 1, w * v.y);
  atomicAdd(o + 2, w * v.z);
  atomicAdd(o + 3, w * v.w);
}

// ---------------- linear + tanh via fp32 WMMA ----------------
// y[n,j] = tanh( sum_k x[n,k] * W[j,k] + b[j] ),  one wave per 16x16 C tile.
// grid = (N_NODES/16) * 2, blockDim = 32 (one wave).
__global__ void k_linear_tanh(const float* __restrict__ xin, const float* __restrict__ W,
                              const float* __restrict__ bias, float* __restrict__ xout,
                              int d) {
  int mt = blockIdx.x >> 1;        // node tile
  int nt = blockIdx.x & 1;         // output-feature tile (32 outputs = 2 tiles)
  int lane = threadIdx.x;
  int half = lane >> 4, l16 = lane & 15;
  long arow = (long)(mt * 16 + l16) * d;       // A row = node, per-lane
  long brow = (long)(nt * 16 + l16) * d;       // B col = output feature j (B[k][j]=W[j][k])
  v8f c = {};
  for (int k0 = 0; k0 < d; k0 += 4) {
    int k = k0 + half * 2;         // lanes 0-15: k0..k0+1, lanes 16-31: k0+2..k0+3
    v2f a, b;
    a[0] = xin[arow + k];  a[1] = xin[arow + k + 1];
    b[0] = W[brow + k];    b[1] = W[brow + k + 1];
    c = __builtin_amdgcn_wmma_f32_16x16x4_f32(false, a, false, b, (short)0, c, false, false);
  }
  int j = nt * 16 + l16;
  float bb = bias[j];
  for (int r = 0; r < 8; ++r) {
    int node = mt * 16 + r + half * 8;          // C rows r (lanes<16) / r+8 (lanes>=16)
    xout[(long)node * HID + j] = tanhf(c[r] + bb);
  }
}

// ---------------- global_sort_pool(k=50) ----------------
// One block per graph. Bitonic sort 512 (500 real + pad) descending on
// h[:, -1] == x3[:, 31], tie-break ascending index (matches stable argsort of -key).
__global__ void k_sort_pool(const float* __restrict__ x1, const float* __restrict__ x2,
                            const float* __restrict__ x3, float* __restrict__ top) {
  __shared__ float sk[512];
  __shared__ int   si[512];
  int g = blockIdx.x, t = threadIdx.x;
  float key = -3.0e38f;
  if (t < NPG) key = x3[((long)g * NPG + t) * HID + 31];
  sk[t] = key; si[t] = t;
  __syncthreads();
  for (int k = 2; k <= 512; k <<= 1) {
    for (int j = k >> 1; j > 0; j >>= 1) {
      int ixj = t ^ j;
      if (ixj > t) {
        float ka = sk[t], kb = sk[ixj];
        int ia = si[t], ib = si[ixj];
        bool beforeAB = (ka > kb) || (ka == kb && ia < ib);  // "a precedes b"
        bool up = ((t & k) == 0);
        if (up ? !beforeAB : beforeAB) {
          sk[t] = kb; sk[ixj] = ka; si[t] = ib; si[ixj] = ia;
        }
      }
      __syncthreads();
    }
  }
  // gather top-50 rows of concat(x1,x2,x3) -> top[g][50][96]
  for (int q = t; q < K_SORT * CAT; q += 512) {
    int i = q / CAT, cch = q - i * CAT;
    long node = (long)g * NPG + si[i];
    float v;
    if (cch < HID)          v = x1[node * HID + cch];
    else if (cch < 2 * HID) v = x2[node * HID + (cch - HID)];
    else                    v = x3[node * HID + (cch - 2 * HID)];
    top[(long)g * (K_SORT * CAT) + q] = v;
  }
}

// ---------------- capsule priors via fp32 WMMA ----------------
// priors[b,o,i,d] = sum_l Wc[o,i,d,l] * top[b,i,l]
// Per (o,i): GEMM M=32(d) x K=96(l) x N=200(b). One wave per 16x16 tile.
// grid = 10 * 50 * 2 * 13, blockDim = 32.
__global__ void k_priors(const float* __restrict__ Wc, const float* __restrict__ top,
                         float* __restrict__ priors) {
  int id = blockIdx.x;
  int nt = id % 13; id /= 13;         // batch tile (13*16 = 208 >= 200)
  int mt = id & 1;  id >>= 1;         // dim tile (2*16 = 32)
  int i  = id % K_SORT; id /= K_SORT;
  int o  = id;                        // < 10
  int lane = threadIdx.x;
  int half = lane >> 4, l16 = lane & 15;
  const float* A = Wc + ((long)o * K_SORT + i) * (HID * CAT);  // [32][96]
  long arow = (long)(mt * 16 + l16) * CAT;
  int n = nt * 16 + l16;                                        // batch index (B column)
  int nc = (n < N_GRAPHS) ? n : (N_GRAPHS - 1);                 // clamp: no divergence
  float mask = (n < N_GRAPHS) ? 1.0f : 0.0f;
  const float* B = top + (long)nc * (K_SORT * CAT) + (long)i * CAT;
  v8f c = {};
  for (int k0 = 0; k0 < CAT; k0 += 4) {
    int k = k0 + half * 2;
    v2f a, b;
    a[0] = A[arow + k];     a[1] = A[arow + k + 1];
    b[0] = B[k] * mask;     b[1] = B[k + 1] * mask;
    c = __builtin_amdgcn_wmma_f32_16x16x4_f32(false, a, false, b, (short)0, c, false, false);
  }
  if (n < N_GRAPHS) {
    long base = ((long)n * NCLASS + o) * (K_SORT * HID) + (long)i * HID;
    for (int r = 0; r < 8; ++r) {
      int m = mt * 16 + r + half * 8;
      priors[base + m] = c[r];
    }
  }
}

// ---------------- dynamic routing (3 iters) + output norm ----------------
// One block per graph. priors[b] is 64KB and the whole tensor fits in L2,
// so re-read from global; only small per-graph state lives in LDS.
__global__ void k_routing(const float* __restrict__ priors, float* __restrict__ out) {
  __shared__ float logit[NCLASS * K_SORT];
  __shared__ float p[NCLASS * K_SORT];
  __shared__ float sv[NCLASS * HID];
  __shared__ float n2s[NCLASS];
  int b = blockIdx.x, t = threadIdx.x;
  const float* pr = priors + (long)b * (NCLASS * K_SORT * HID);
  if (t < NCLASS * K_SORT) logit[t] = 0.0f;
  __syncthreads();
  for (int it = 0; it < 3; ++it) {
    // softmax over o for each i
    if (t < K_SORT) {
      float m = -3.0e38f;
      for (int o = 0; o < NCLASS; ++o) m = fmaxf(m, logit[o * K_SORT + t]);
      float e[NCLASS]; float ssum = 0.0f;
      for (int o = 0; o < NCLASS; ++o) { e[o] = expf(logit[o * K_SORT + t] - m); ssum += e[o]; }
      float inv = 1.0f / ssum;
      for (int o = 0; o < NCLASS; ++o) p[o * K_SORT + t] = e[o] * inv;
    }
    __syncthreads();
    // s[o,d] = sum_i p[o,i] * priors[o,i,d]
    if (t < NCLASS * HID) {
      int o = t >> 5, dd = t & 31;
      float s = 0.0f;
      for (int i = 0; i < K_SORT; ++i)
        s += p[o * K_SORT + i] * pr[(long)o * (K_SORT * HID) + i * HID + dd];
      sv[t] = s;
    }
    __syncthreads();
    if (t < NCLASS) {
      float n2 = 0.0f;
      for (int dd = 0; dd < HID; ++dd) { float v = sv[t * HID + dd]; n2 += v * v; }
      n2s[t] = n2;
    }
    __syncthreads();
    if (it < 2) {
      // v = squash(s);  logits += <priors, v>
      if (t < NCLASS * HID) {
        int o = t >> 5;
        float n2 = n2s[o];
        float scale = n2 / ((1.0f + n2) * sqrtf(n2));
        sv[t] *= scale;
      }
      __syncthreads();
      if (t < NCLASS * K_SORT) {
        int o = t / K_SORT, i = t - o * K_SORT;
        float dot = 0.0f;
        for (int dd = 0; dd < HID; ++dd)
          dot += pr[(long)o * (K_SORT * HID) + i * HID + dd] * sv[o * HID + dd];
        logit[t] += dot;
      }
      __syncthreads();
    }
  }
  // ||squash(s)|| = n2 / (1 + n2)
  if (t < NCLASS) out[(long)b * NCLASS + t] = n2s[t] / (1.0f + n2s[t]);
}

// ---------------- host orchestration ----------------
extern "C" void kernel_launch(void* const* d_in, const int* in_sizes, int n_in,
                              void* d_out, int out_size, void* d_ws, size_t ws_size,
                              hipStream_t stream) {
  (void)in_sizes; (void)n_in; (void)out_size; (void)ws_size;
  const float* x  = (const float*)d_in[0];
  const int*   ei = (const int*)d_in[1];
  const int* src = ei;
  const int* dst = ei + N_EDGES;
  // d_in[2] = batch (unused: graphs are contiguous blocks of 500)
  const float* W1 = (const float*)d_in[3]; const float* b1 = (const float*)d_in[4];
  const float* W2 = (const float*)d_in[5]; const float* b2 = (const float*)d_in[6];
  const float* W3 = (const float*)d_in[7]; const float* b3 = (const float*)d_in[8];
  const float* Wc = (const float*)d_in[9];

  float* ws   = (float*)d_ws;
  float* dinv = ws;                                   // N
  float* bufA = dinv + N_NODES;                       // N*64
  float* bufB = bufA + (size_t)N_NODES * NFEAT;       // N*64
  float* x1   = bufB + (size_t)N_NODES * NFEAT;       // N*32
  float* x2   = x1 + (size_t)N_NODES * HID;           // N*32
  float* x3   = x2 + (size_t)N_NODES * HID;           // N*32
  float* top    = bufA;   // reused after conv stack: 200*50*96
  float* priors = bufB;   // reused: 200*10*50*32

  k_init_deg<<<(N_NODES + 255) / 256, 256, 0, stream>>>(dinv);
  k_deg_edge<<<(N_EDGES + 255) / 256, 256, 0, stream>>>(dinv, dst);
  k_dinv<<<(N_NODES + 255) / 256, 256, 0, stream>>>(dinv);

  struct Layer { const float* xin; int d; const float* W; const float* bias; float* xout; };
  Layer L[3] = { { x,  NFEAT, W1, b1, x1 },
                 { x1, HID,   W2, b2, x2 },
                 { x2, HID,   W3, b3, x3 } };
  for (int l = 0; l < 3; ++l) {
    const float* cur = L[l].xin;
    int d = L[l].d;
    for (int step = 0; step < 3; ++step) {
      float* nxt = (cur == bufA) ? bufB : bufA;
      int nd = N_NODES * d;
      k_prop_init<<<(nd + 255) / 256, 256, 0, stream>>>(cur, nxt, dinv, d);
      int ed = N_EDGES * (d / 4);
      k_prop_edge<<<(ed + 255) / 256, 256, 0, stream>>>(cur, nxt, src, dst, dinv, d / 4);
      cur = nxt;
    }
    k_linear_tanh<<<(N_NODES / 16) * 2, 32, 0, stream>>>(cur, L[l].W, L[l].bias, L[l].xout, d);
  }

  k_sort_pool<<<N_GRAPHS, 512, 0, stream>>>(x1, x2, x3, top);
  k_priors<<<NCLASS * K_SORT * 2 * 13, 32, 0, stream>>>(Wc, top, priors);
  k_routing<<<N_GRAPHS, 512, 0, stream>>>(priors, (float*)d_out);
}